// MF_11905649344965
// MI455X (gfx1250) — compile-verified
//
#include <hip/hip_runtime.h>

// 3x3 median blur ([16,3,512,512] fp32, zero padding applied to the
// pre-scaled value t=(v+1)/2, output = median*2-1) + mask passthrough.
// Median kernel: register-sliding 4x4 tile/thread, b128 coalesced, HBM-bound.
// Mask copy: CDNA5 async-to-LDS bulk copy (global_load_async_to_lds_b128 /
// global_store_async_from_lds_b128 + s_wait_asynccnt).

constexpr int W  = 512;
constexpr int H  = 512;
constexpr int TX = 4;              // columns per thread (one float4)
constexpr int TY = 4;              // rows per thread
constexpr int CT = W / TX;         // 128 column tiles per row
constexpr int RT = H / TY;         // 128 row tiles per plane

__device__ __forceinline__ void cas(float& a, float& b) {
    float lo = fminf(a, b);
    float hi = fmaxf(a, b);
    a = lo; b = hi;
}

// Devillard opt_med9: exact median of 9 via 19 compare-exchanges.
// Unused halves of the CASes are dead-code-eliminated; the compiler also
// CSEs the shared sub-networks between the 4 overlapping windows.
__device__ __forceinline__ float med9(float p0, float p1, float p2,
                                      float p3, float p4, float p5,
                                      float p6, float p7, float p8) {
    cas(p1, p2); cas(p4, p5); cas(p7, p8);
    cas(p0, p1); cas(p3, p4); cas(p6, p7);
    cas(p1, p2); cas(p4, p5); cas(p7, p8);
    cas(p0, p3); cas(p5, p8); cas(p4, p7);
    cas(p3, p6); cas(p1, p4); cas(p2, p5);
    cas(p4, p7); cas(p4, p2); cas(p6, p4);
    cas(p4, p2);
    return p4;
}

// Load 6 pre-scaled values (cols x0-1 .. x0+4) of row y into a[0..5].
// Out-of-bounds (zero padding in pre-scaled space) -> 0.0f.
__device__ __forceinline__ void load_row(const float* __restrict__ src,
                                         int y, int x0,
                                         bool has_left, bool has_right,
                                         float a[6]) {
    if (y < 0 || y >= H) {
        #pragma unroll
        for (int i = 0; i < 6; ++i) a[i] = 0.0f;
        return;
    }
    const float* row = src + (size_t)y * W + x0;
    float4 v = *reinterpret_cast<const float4*>(row);   // 16B-aligned b128 load
    a[1] = fmaf(v.x, 0.5f, 0.5f);
    a[2] = fmaf(v.y, 0.5f, 0.5f);
    a[3] = fmaf(v.z, 0.5f, 0.5f);
    a[4] = fmaf(v.w, 0.5f, 0.5f);
    a[0] = has_left  ? fmaf(row[-1], 0.5f, 0.5f) : 0.0f;
    a[5] = has_right ? fmaf(row[4],  0.5f, 0.5f) : 0.0f;
}

// blockDim = (128, 2, 1): threadIdx.x = column tile (coalesced within wave32),
// blockIdx.y*2+threadIdx.y = row tile, blockIdx.z = plane. No div/mod.
__global__ __launch_bounds__(256)
void median3x3_kernel(const float* __restrict__ img,
                      float* __restrict__ out) {
    const int tx    = threadIdx.x;
    const int ty    = blockIdx.y * blockDim.y + threadIdx.y;
    const int plane = blockIdx.z;

    const float* src = img + (size_t)plane * H * W;
    float*       dst = out + (size_t)plane * H * W;

    const int x0 = tx * TX;
    const int y0 = ty * TY;
    const bool has_left  = (x0 > 0);
    const bool has_right = (x0 + TX < W);

    float r0[6], r1[6], r2[6];
    load_row(src, y0 - 1, x0, has_left, has_right, r0);
    load_row(src, y0,     x0, has_left, has_right, r1);

    #pragma unroll
    for (int i = 0; i < TY; ++i) {
        const int y = y0 + i;

        // gfx1250 global_prefetch_b8: pull the row loaded in the next iteration.
        if (y + 2 < H) {
            __builtin_prefetch(src + (size_t)(y + 2) * W + x0, 0, 0);
        }

        load_row(src, y + 1, x0, has_left, has_right, r2);

        float4 o;
        o.x = fmaf(med9(r0[0], r0[1], r0[2],
                        r1[0], r1[1], r1[2],
                        r2[0], r2[1], r2[2]), 2.0f, -1.0f);
        o.y = fmaf(med9(r0[1], r0[2], r0[3],
                        r1[1], r1[2], r1[3],
                        r2[1], r2[2], r2[3]), 2.0f, -1.0f);
        o.z = fmaf(med9(r0[2], r0[3], r0[4],
                        r1[2], r1[3], r1[4],
                        r2[2], r2[3], r2[4]), 2.0f, -1.0f);
        o.w = fmaf(med9(r0[3], r0[4], r0[5],
                        r1[3], r1[4], r1[5],
                        r2[3], r2[4], r2[5]), 2.0f, -1.0f);

        *reinterpret_cast<float4*>(dst + (size_t)y * W + x0) = o;  // b128 store

        #pragma unroll
        for (int k = 0; k < 6; ++k) { r0[k] = r1[k]; r1[k] = r2[k]; }
    }
}

// ---------------------------------------------------------------------------
// Mask passthrough via the CDNA5 async-to-LDS engine.
// Each thread: 8 outstanding b128 async loads into its private 128B LDS slot
// (32KB/block in flight), one s_wait_asynccnt, 8 b128 async stores.
// Each k-step is a fully coalesced device-wide sweep.
// ---------------------------------------------------------------------------
constexpr int CPB = 8;  // 16-byte chunks per thread

typedef __attribute__((address_space(3))) unsigned char lds_byte_t;

__global__ __launch_bounds__(256)
void mask_copy_async_kernel(const float* __restrict__ src,
                            float* __restrict__ dst,
                            int nchunks) {
    __shared__ __align__(16) unsigned char smem[256 * CPB * 16];

    const int tid    = blockIdx.x * blockDim.x + threadIdx.x;
    const int stride = gridDim.x * blockDim.x;

    // True LDS byte offset of this thread's slot (generic -> AS(3) -> int).
    const unsigned lds_base =
        (unsigned)(unsigned long long)(lds_byte_t*)&smem[threadIdx.x * (CPB * 16)];

    #pragma unroll
    for (int k = 0; k < CPB; ++k) {
        const int c = tid + k * stride;
        if (c < nchunks) {
            const unsigned voff  = (unsigned)c * 16u;          // global byte offset
            const unsigned laddr = lds_base + (unsigned)k * 16u;
            asm volatile("global_load_async_to_lds_b128 %0, %1, %2"
                         :: "v"(laddr), "v"(voff), "s"(src)
                         : "memory");
        }
    }

    // Wait for all 8 async loads of this wave to land in LDS.
    asm volatile("s_wait_asynccnt 0x0" ::: "memory");

    #pragma unroll
    for (int k = 0; k < CPB; ++k) {
        const int c = tid + k * stride;
        if (c < nchunks) {
            const unsigned voff  = (unsigned)c * 16u;
            const unsigned laddr = lds_base + (unsigned)k * 16u;
            asm volatile("global_store_async_from_lds_b128 %0, %1, %2"
                         :: "v"(voff), "v"(laddr), "s"(dst)
                         : "memory");
        }
    }
    // S_ENDPGM performs an implicit wait-idle, draining ASYNCcnt.
}

extern "C" void kernel_launch(void* const* d_in, const int* in_sizes, int n_in,
                              void* d_out, int out_size, void* d_ws, size_t ws_size,
                              hipStream_t stream) {
    const float* img  = (const float*)d_in[0];   // [16,3,512,512] fp32
    const float* mask = (const float*)d_in[1];   // [16,1,512,512] fp32
    float*       out  = (float*)d_out;           // filtered ‖ mask

    const int planes = in_sizes[0] / (H * W);    // 48

    dim3 mblock(CT, 2, 1);                       // 128 x 2 = 256 threads
    dim3 mgrid(1, RT / 2, planes);               // 64 row-tile pairs x 48 planes
    median3x3_kernel<<<mgrid, mblock, 0, stream>>>(img, out);

    // Mask: async-LDS bulk copy. 16-byte chunks.
    const int nchunks = in_sizes[1] / 4;         // 4 floats per chunk
    const int cthreads = 256;
    const int cblocks  = (nchunks + cthreads * CPB - 1) / (cthreads * CPB); // 512
    mask_copy_async_kernel<<<cblocks, cthreads, 0, stream>>>(
        mask, out + (size_t)in_sizes[0], nchunks);
}